// GCNLayer_69415261437959
// MI455X (gfx1250) — compile-verified
//
#include <hip/hip_runtime.h>

#define N_NODES 100000
#define N_EDGES 1600000
#define DIM 64
#define LN_EPS 1e-5f

typedef __attribute__((ext_vector_type(2))) float v2f;
typedef __attribute__((ext_vector_type(8))) float v8f;

// ---------------------------------------------------------------------------
// Kernel 1: zero the aggregation workspace
// ---------------------------------------------------------------------------
__global__ void gcn_zero_kernel(float* __restrict__ p, int n) {
    int i = blockIdx.x * blockDim.x + threadIdx.x;
    int stride = gridDim.x * blockDim.x;
    for (; i < n; i += stride) p[i] = 0.0f;
}

// ---------------------------------------------------------------------------
// Kernel 2: SpMM  agg[r] += val * x[c]
// 16 lanes per edge, each lane owns 4 contiguous dims (float4 gather),
// accumulate with native f32 global atomics (L2-resident working set).
// ---------------------------------------------------------------------------
__global__ void gcn_spmm_kernel(const float* __restrict__ x,
                                const int*   __restrict__ erow,
                                const int*   __restrict__ ecol,
                                const float* __restrict__ eval,
                                float*       __restrict__ agg) {
    const int tid    = blockIdx.x * blockDim.x + threadIdx.x;
    const int lane16 = tid & 15;
    int e            = tid >> 4;
    const int estr   = (gridDim.x * blockDim.x) >> 4;
    for (; e < N_EDGES; e += estr) {
        const int   r = erow[e];
        const int   c = ecol[e];
        const float v = eval[e];
        const float4 xv = *reinterpret_cast<const float4*>(x + c * DIM + lane16 * 4);
        float* dst = agg + r * DIM + lane16 * 4;
        unsafeAtomicAdd(dst + 0, v * xv.x);
        unsafeAtomicAdd(dst + 1, v * xv.y);
        unsafeAtomicAdd(dst + 2, v * xv.z);
        unsafeAtomicAdd(dst + 3, v * xv.w);
    }
}

// ---------------------------------------------------------------------------
// Kernel 3: fused  h = agg @ W^T + b ; LayerNorm(h) ; ReLU
// One wave computes a 16x64 output tile with V_WMMA_F32_16X16X4_F32.
//   A (16x4 f32):  lanes 0-15 -> M=lane, VGPRs = K=0,1 ; lanes 16-31 -> K=2,3
//   B (4x16 f32):  lanes 0-15 -> N=lane, VGPRs = K=0,1 ; lanes 16-31 -> K=2,3
//   C/D (16x16):   VGPR g holds row g (lanes 0-15) / row g+8 (lanes 16-31)
// LayerNorm reduces across n: 4 accumulator adds + shfl_xor(1,2,4,8), which
// stays within each 16-lane half of the wave32.
// ---------------------------------------------------------------------------
__global__ void gcn_gemm_ln_kernel(const float* __restrict__ agg,
                                   const float* __restrict__ W,
                                   const float* __restrict__ bias,
                                   const float* __restrict__ gamma,
                                   const float* __restrict__ beta,
                                   float*       __restrict__ out) {
    const int lane    = threadIdx.x & 31;
    const int wave    = (blockIdx.x * blockDim.x + threadIdx.x) >> 5;
    const int rowbase = wave * 16;
    if (rowbase >= N_NODES) return;   // uniform per wave: EXEC stays all-ones

    const int m  = lane & 15;   // row within A-tile / column within N-tile
    const int kh = lane >> 4;   // K-half select for A/B fragments

    // Pull W (16 KB) toward the caches.
    __builtin_prefetch(W + (threadIdx.x << 4), 0, 0);

    v8f acc0 = {}, acc1 = {}, acc2 = {}, acc3 = {};

    const float* arow = agg + (rowbase + m) * DIM;
    const float* w0   = W + (m +  0) * DIM;
    const float* w1   = W + (m + 16) * DIM;
    const float* w2   = W + (m + 32) * DIM;
    const float* w3   = W + (m + 48) * DIM;

#pragma unroll
    for (int ks = 0; ks < 16; ++ks) {
        const int kk = ks * 4 + kh * 2;   // even -> 8B aligned float2 loads
        const v2f a  = *reinterpret_cast<const v2f*>(arow + kk);
        const v2f b0 = *reinterpret_cast<const v2f*>(w0 + kk);  // B[k][n] = W[n][k]
        const v2f b1 = *reinterpret_cast<const v2f*>(w1 + kk);
        const v2f b2 = *reinterpret_cast<const v2f*>(w2 + kk);
        const v2f b3 = *reinterpret_cast<const v2f*>(w3 + kk);
        acc0 = __builtin_amdgcn_wmma_f32_16x16x4_f32(false, a, false, b0, (short)0, acc0, false, false);
        acc1 = __builtin_amdgcn_wmma_f32_16x16x4_f32(false, a, false, b1, (short)0, acc1, false, false);
        acc2 = __builtin_amdgcn_wmma_f32_16x16x4_f32(false, a, false, b2, (short)0, acc2, false, false);
        ac3:;
        acc3 = __builtin_amdgcn_wmma_f32_16x16x4_f32(false, a, false, b3, (short)0, acc3, false, false);
    }

    // Per-lane epilogue constants for the 4 N-tiles (n = nt*16 + m).
    float bia[4], gam[4], bet[4];
#pragma unroll
    for (int nt = 0; nt < 4; ++nt) {
        const int n = nt * 16 + m;
        bia[nt] = bias[n];
        gam[nt] = gamma[n];
        bet[nt] = beta[n];
    }

#pragma unroll
    for (int g = 0; g < 8; ++g) {
        float v0 = acc0[g] + bia[0];
        float v1 = acc1[g] + bia[1];
        float v2 = acc2[g] + bia[2];
        float v3 = acc3[g] + bia[3];

        float s1 = v0 + v1 + v2 + v3;
        float s2 = v0 * v0 + v1 * v1 + v2 * v2 + v3 * v3;
#pragma unroll
        for (int xm = 1; xm <= 8; xm <<= 1) {   // stays within 16-lane half
            s1 += __shfl_xor(s1, xm, 32);
            s2 += __shfl_xor(s2, xm, 32);
        }
        const float mu   = s1 * (1.0f / 64.0f);
        const float var  = s2 * (1.0f / 64.0f) - mu * mu;
        const float rstd = rsqrtf(var + LN_EPS);

        const int r = rowbase + (kh ? g + 8 : g);
        float* orow = out + r * DIM + m;
        orow[ 0] = fmaxf((v0 - mu) * rstd * gam[0] + bet[0], 0.0f);
        orow[16] = fmaxf((v1 - mu) * rstd * gam[1] + bet[1], 0.0f);
        orow[32] = fmaxf((v2 - mu) * rstd * gam[2] + bet[2], 0.0f);
        orow[48] = fmaxf((v3 - mu) * rstd * gam[3] + bet[3], 0.0f);
    }
}

// ---------------------------------------------------------------------------
extern "C" void kernel_launch(void* const* d_in, const int* in_sizes, int n_in,
                              void* d_out, int out_size, void* d_ws, size_t ws_size,
                              hipStream_t stream) {
    const float* x     = (const float*)d_in[0];
    const int*   erow  = (const int*)  d_in[1];
    const int*   ecol  = (const int*)  d_in[2];
    const float* eval  = (const float*)d_in[3];
    const float* W     = (const float*)d_in[4];
    const float* bias  = (const float*)d_in[5];
    const float* gamma = (const float*)d_in[6];
    const float* beta  = (const float*)d_in[7];
    float* out = (float*)d_out;
    float* agg = (float*)d_ws;   // N_NODES * DIM floats = 25.6 MB scratch

    gcn_zero_kernel<<<4096, 256, 0, stream>>>(agg, N_NODES * DIM);
    gcn_spmm_kernel<<<8192, 256, 0, stream>>>(x, erow, ecol, eval, agg);

    const int waves_needed  = N_NODES / 16;          // 6250 row tiles
    const int waves_per_blk = 8;                     // 256 threads
    const int blocks        = (waves_needed + waves_per_blk - 1) / waves_per_blk;
    gcn_gemm_ln_kernel<<<blocks, 256, 0, stream>>>(agg, W, bias, gamma, beta, out);
}